// SumArray_78108275245110
// MI455X (gfx1250) — compile-verified
//
#include <hip/hip_runtime.h>

typedef __attribute__((ext_vector_type(2))) float v2f;
typedef __attribute__((ext_vector_type(4))) float v4f;
typedef __attribute__((ext_vector_type(8))) float v8f;

// Wave32 reduction of two per-lane fp32 accumulators using one
// V_WMMA_F32_16X16X4_F32:  D = A(16x4, lane accumulators) * B(4x16, ones) + 0.
// A layout: lane m%16 holds A[m][k] with k selected by VGPR index and lane
// half, so (a0, a1) per lane covers all 64 values. D[m][n] = rowsum(m)
// replicated across columns; each lane sums its 8 D VGPRs, giving lanes 0-15
// sum(rows 0..7) and lanes 16-31 sum(rows 8..15); xor-16 completes the wave.
// EXEC must be all ones here (no divergence around the call).
__device__ __forceinline__ float wave_reduce_wmma(float a0, float a1) {
  v2f a; a[0] = a0; a[1] = a1;
  v2f b; b[0] = 1.0f; b[1] = 1.0f;
  v8f c = {};
  v8f d = __builtin_amdgcn_wmma_f32_16x16x4_f32(
      /*neg_a=*/false, a, /*neg_b=*/false, b,
      /*c_mod=*/(short)0, c, /*reuse_a=*/false, /*reuse_b=*/false);
  float s = ((d[0] + d[1]) + (d[2] + d[3])) + ((d[4] + d[5]) + (d[6] + d[7]));
  s += __shfl_xor(s, 16, 32);  // combine the two half-wave row groups
  return s;
}

// Pass 1: streaming pass. 4-way unrolled grid-stride loop -> a clause of four
// independent global_load_b128 per iteration (4 x 512 B outstanding per wave)
// to hide HBM latency; one fp32 partial per block.
__global__ void __launch_bounds__(256)
sum_partial_kernel(const float* __restrict__ x, float* __restrict__ partials,
                   int n) {
  const int tid  = threadIdx.x;
  const int gidx = blockIdx.x * blockDim.x + tid;
  const int gsz  = gridDim.x * blockDim.x;

  const v4f* __restrict__ x4 = (const v4f*)x;
  const int n4 = n >> 2;

  float s0 = 0.0f, s1 = 0.0f, s2 = 0.0f, s3 = 0.0f;
  int i = gidx;
  // main unrolled loop: 4 independent load streams
  for (; i + 3 * gsz < n4; i += 4 * gsz) {
    v4f v0 = x4[i];
    v4f v1 = x4[i + gsz];
    v4f v2 = x4[i + 2 * gsz];
    v4f v3 = x4[i + 3 * gsz];
    s0 += (v0[0] + v0[1]) + (v0[2] + v0[3]);
    s1 += (v1[0] + v1[1]) + (v1[2] + v1[3]);
    s2 += (v2[0] + v2[1]) + (v2[2] + v2[3]);
    s3 += (v3[0] + v3[1]) + (v3[2] + v3[3]);
  }
  // remainder vector iterations
  for (; i < n4; i += gsz) {
    v4f v = x4[i];
    s0 += (v[0] + v[1]) + (v[2] + v[3]);
  }
  float a0 = s0 + s2;
  float a1 = s1 + s3;
  // scalar tail (n % 4), handled once by a single thread
  if (gidx == 0) {
    for (int j = n4 << 2; j < n; ++j) a0 += x[j];
  }

  // wave32 reduction via WMMA (uniform control flow: EXEC all ones)
  float ws = wave_reduce_wmma(a0, a1);

  __shared__ float wsums[8];            // 256 threads = 8 waves
  const int wave = tid >> 5;
  if ((tid & 31) == 0) wsums[wave] = ws;
  __syncthreads();
  if (tid == 0) {
    float t = 0.0f;
#pragma unroll
    for (int w = 0; w < 8; ++w) t += wsums[w];
    partials[blockIdx.x] = t;
  }
}

// Pass 2: one block folds all block partials into d_out[0] (plain store,
// fixed tree order -> bit-deterministic across graph replays).
__global__ void __launch_bounds__(256)
sum_final_kernel(const float* __restrict__ partials, float* __restrict__ out,
                 int np) {
  const int tid = threadIdx.x;
  float a0 = 0.0f, a1 = 0.0f;
  for (int i = tid; i < np; i += 512) {
    a0 += partials[i];
    if (i + 256 < np) a1 += partials[i + 256];
  }

  float ws = wave_reduce_wmma(a0, a1);

  __shared__ float wsums[8];
  const int wave = tid >> 5;
  if ((tid & 31) == 0) wsums[wave] = ws;
  __syncthreads();
  if (tid == 0) {
    float t = 0.0f;
#pragma unroll
    for (int w = 0; w < 8; ++w) t += wsums[w];
    out[0] = t;
  }
}

extern "C" void kernel_launch(void* const* d_in, const int* in_sizes, int n_in,
                              void* d_out, int out_size, void* d_ws,
                              size_t ws_size, hipStream_t stream) {
  const float* x   = (const float*)d_in[0];
  float* out       = (float*)d_out;
  float* partials  = (float*)d_ws;
  const int n      = in_sizes[0];

  int nblocks = 2048;  // 16K wave32 waves in flight: saturates 23.3 TB/s HBM
  const size_t need = (size_t)nblocks * sizeof(float);
  if (ws_size < need) nblocks = (int)(ws_size / sizeof(float));
  if (nblocks < 1) nblocks = 1;

  sum_partial_kernel<<<nblocks, 256, 0, stream>>>(x, partials, n);
  sum_final_kernel<<<1, 256, 0, stream>>>(partials, out, nblocks);
}